// AttShiftW_21414706938552
// MI455X (gfx1250) — compile-verified
//
#include <hip/hip_runtime.h>
#include <math.h>

#define Bn 16
#define Cn 64
#define Hn 128
#define Wn 256
#define HWn (Hn * Wn)          /* 32768 */
#define Nn  (Bn * HWn)         /* 524288 */
#define PI_F 3.14159265358979323846f

typedef float v2f __attribute__((ext_vector_type(2)));
typedef float v8f __attribute__((ext_vector_type(8)));

// ---------------------------------------------------------------------------
// K1: channel mean via V_WMMA_F32_16X16X4_F32.  D = A(ones/64, 16x4) * B(4x16)
// accumulated over 16 k-steps (C=64).  One wave -> 16 consecutive hw outputs.
// B layout (32-bit, 4x16): lane L<16 holds (N=L, K=0 in v0, K=1 in v1);
// lane L>=16 holds (N=L-16, K=2 in v0, K=3 in v1).  D row M=0 = acc[0], lanes 0-15.
// ---------------------------------------------------------------------------
__global__ __launch_bounds__(256) void k_cmean_wmma(const float* __restrict__ x,
                                                    float* __restrict__ out) {
    int gtid = blockIdx.x * 256 + threadIdx.x;
    int wave = gtid >> 5;                 // 32768 waves total
    int lane = threadIdx.x & 31;
    int b    = wave / (HWn / 16);
    int tile = wave % (HWn / 16);
    int base = tile * 16;
    int col  = lane & 15;
    int kp   = (lane >> 4) << 1;          // 0 or 2
    const float* xb = x + (size_t)b * Cn * HWn + base + col;

    v2f a; a.x = 1.0f / 64.0f; a.y = 1.0f / 64.0f;
    v8f acc = {};
#pragma unroll
    for (int k0 = 0; k0 < Cn; k0 += 4) {
        v2f bm;
        bm.x = xb[(size_t)(k0 + kp) * HWn];
        bm.y = xb[(size_t)(k0 + kp + 1) * HWn];
        acc = __builtin_amdgcn_wmma_f32_16x16x4_f32(false, a, false, bm,
                                                    (short)0, acc, false, false);
    }
    if (lane < 16) out[(size_t)b * HWn + base + lane] = acc[0];
}

// K2: per-batch min/max over the 32768-element mean map (one block per b).
__global__ __launch_bounds__(256) void k_minmax(const float* __restrict__ cm,
                                                float* __restrict__ mnmx) {
    __shared__ float smn[256], smx[256];
    int b = blockIdx.x;
    const float* p = cm + (size_t)b * HWn;
    float mn = 3.4e38f, mx = -3.4e38f;
    for (int i = threadIdx.x; i < HWn; i += 256) {
        float v = p[i];
        mn = fminf(mn, v);
        mx = fmaxf(mx, v);
    }
    smn[threadIdx.x] = mn; smx[threadIdx.x] = mx;
    __syncthreads();
    for (int s = 128; s > 0; s >>= 1) {
        if (threadIdx.x < s) {
            smn[threadIdx.x] = fminf(smn[threadIdx.x], smn[threadIdx.x + s]);
            smx[threadIdx.x] = fmaxf(smx[threadIdx.x], smx[threadIdx.x + s]);
        }
        __syncthreads();
    }
    if (threadIdx.x == 0) { mnmx[2 * b] = smn[0]; mnmx[2 * b + 1] = smx[0]; }
}

// K3: normalize in place; init union-find parents (mask = norm >= 0.4).
__global__ __launch_bounds__(256) void k_norm_init(float* __restrict__ cm,
                                                   const float* __restrict__ mnmx,
                                                   int* __restrict__ parent) {
    int i = blockIdx.x * 256 + threadIdx.x;       // exactly Nn threads
    int b = i / HWn;
    float mn = mnmx[2 * b], mx = mnmx[2 * b + 1];
    float v = (cm[i] - mn) / (mx - mn);
    cm[i] = v;
    parent[i] = (v >= 0.4f) ? i : -1;
}

// ---------------------------------------------------------------------------
// Union-find (lo root linked under hi root -> final root = max linear index
// in component, exactly matching the reference's label semantics).
// ---------------------------------------------------------------------------
__device__ __forceinline__ int find_root(const int* P, int i) {
    const volatile int* vp = (const volatile int*)P;
    int p = vp[i];
    while (p != i) { i = p; p = vp[i]; }
    return i;
}

__device__ void unite(int* P, int a, int b) {
    int ra = find_root(P, a);
    int rb = find_root(P, b);
    while (ra != rb) {
        int hi = ra > rb ? ra : rb;
        int lo = ra > rb ? rb : ra;
        int old = atomicCAS(&P[lo], lo, hi);
        if (old == lo) return;
        ra = find_root(P, old);
        rb = find_root(P, hi);
    }
}

// K4: hook along +W, +H, +B (6-connectivity; symmetric via union).
__global__ __launch_bounds__(256) void k_union(int* __restrict__ parent) {
    int i = blockIdx.x * 256 + threadIdx.x;
    if (parent[i] < 0) return;
    int w = i % Wn;
    int h = (i / Wn) % Hn;
    if (w + 1 < Wn  && parent[i + 1]   >= 0) unite(parent, i, i + 1);
    if (h + 1 < Hn  && parent[i + Wn]  >= 0) unite(parent, i, i + Wn);
    if (i + HWn < Nn && parent[i + HWn] >= 0) unite(parent, i, i + HWn);
}

// K5: path-compress to root.
__global__ __launch_bounds__(256) void k_flatten(int* __restrict__ parent) {
    int i = blockIdx.x * 256 + threadIdx.x;
    if (parent[i] >= 0) parent[i] = find_root(parent, i);
}

// zero counts buffer (poisoned on first call).
__global__ __launch_bounds__(256) void k_zero_counts(int* __restrict__ counts) {
    int i = blockIdx.x * 256 + threadIdx.x;
    if (i <= Nn) counts[i] = 0;
}

// zero per-input argmax keys + centroid stats.
__global__ void k_zero_small(unsigned long long* __restrict__ bestkey,
                             int* __restrict__ stats) {
    int t = threadIdx.x;
    if (t < 16) bestkey[t] = 0ull;
    if (t < 48) stats[t] = 0;
}

// K6a/b/c: per-slice histogram, order-independent argmax, touched-entry reset.
__global__ __launch_bounds__(256) void k_count(const int* __restrict__ parent,
                                               int* __restrict__ counts, int b) {
    int i = b * HWn + blockIdx.x * 256 + threadIdx.x;
    int p = parent[i];
    if (p >= 0) atomicAdd(&counts[p + 1], 1);
}

__global__ __launch_bounds__(256) void k_argmax(const int* __restrict__ parent,
                                                const int* __restrict__ counts,
                                                unsigned long long* __restrict__ bestkey,
                                                int b) {
    int i = b * HWn + blockIdx.x * 256 + threadIdx.x;
    int p = parent[i];
    if (p >= 0) {
        unsigned int lab = (unsigned int)(p + 1);
        unsigned long long key =
            ((unsigned long long)(unsigned int)counts[lab] << 32) | (0xFFFFFFFFu - lab);
        atomicMax(&bestkey[b], key);   // max count, tie -> smallest label
    }
}

__global__ __launch_bounds__(256) void k_reset(const int* __restrict__ parent,
                                               int* __restrict__ counts, int b) {
    int i = b * HWn + blockIdx.x * 256 + threadIdx.x;
    int p = parent[i];
    if (p >= 0) counts[p + 1] = 0;
}

// K7: zero non-largest-region values in place; integer centroid accumulation
// (block-reduced, then 3 int atomics per block -> deterministic & exact).
__global__ __launch_bounds__(256) void k_big(const int* __restrict__ parent,
                                             const unsigned long long* __restrict__ bestkey,
                                             float* __restrict__ outR,
                                             int* __restrict__ stats) {
    __shared__ int sc[256], shh[256], sww[256];
    int i = blockIdx.x * 256 + threadIdx.x;
    int b = i / HWn;
    int p = parent[i];
    unsigned int bl = 0xFFFFFFFFu - (unsigned int)(bestkey[b] & 0xFFFFFFFFull);
    int c = 0, hv = 0, wv = 0;
    if (p >= 0 && (unsigned int)(p + 1) == bl) {
        int rem = i - b * HWn;
        int hh = rem / Wn;
        c = 1; hv = hh; wv = rem - hh * Wn;
    } else {
        outR[i] = 0.0f;
    }
    sc[threadIdx.x] = c; shh[threadIdx.x] = hv; sww[threadIdx.x] = wv;
    __syncthreads();
    for (int s = 128; s > 0; s >>= 1) {
        if (threadIdx.x < s) {
            sc[threadIdx.x]  += sc[threadIdx.x + s];
            shh[threadIdx.x] += shh[threadIdx.x + s];
            sww[threadIdx.x] += sww[threadIdx.x + s];
        }
        __syncthreads();
    }
    if (threadIdx.x == 0 && sc[0]) {
        atomicAdd(&stats[b], sc[0]);
        atomicAdd(&stats[16 + b], shh[0]);
        atomicAdd(&stats[32 + b], sww[0]);
    }
}

__device__ __forceinline__ float sph_dist(float t1, float p1, float t2, float p2) {
    float cosd = sinf(t1) * sinf(t2) + cosf(t1) * cosf(t2) * cosf(p1 - p2);
    float d = acosf(cosd) / PI_F;
    return (d != d) ? 0.0f : d;            // nan_to_num
}

// K8: angles + spherical distances -> w1, w2.
__global__ void k_final(const int* __restrict__ stats,   // [3][48]
                        float* __restrict__ w1, float* __restrict__ w2) {
    int b = threadIdx.x;
    if (b >= 16) return;
    float th[3], ph[3];
    for (int i = 0; i < 3; i++) {
        const int* s = stats + i * 48;
        float cnt = (float)s[b];
        float hm = (float)s[16 + b] / cnt;   // 0/0 -> NaN, as in reference
        float wm = (float)s[32 + b] / cnt;
        ph[i] = (0.5f - hm / (float)Hn) * PI_F;
        th[i] = (wm / (float)Wn - 0.5f) * 2.0f * PI_F;
    }
    w1[b] = sph_dist(th[0], ph[0], th[1], ph[1]);
    w2[b] = sph_dist(th[1], ph[1], th[2], ph[2]);
}

extern "C" void kernel_launch(void* const* d_in, const int* in_sizes, int n_in,
                              void* d_out, int out_size, void* d_ws, size_t ws_size,
                              hipStream_t stream) {
    (void)in_sizes; (void)n_in; (void)out_size; (void)ws_size;
    float* out = (float*)d_out;
    char* ws = (char*)d_ws;

    // workspace layout (~4.2 MB)
    int* parent = (int*)(ws);                                  // Nn ints
    int* counts = (int*)(ws + (size_t)Nn * 4);                 // Nn+1 ints
    float* mnmx = (float*)(ws + (size_t)Nn * 4 + 2097160);     // 32 floats
    unsigned long long* bestkey =
        (unsigned long long*)(ws + (size_t)Nn * 4 + 2097160 + 128);   // 3*16 u64
    int* stats = (int*)(ws + (size_t)Nn * 4 + 2097160 + 128 + 384);   // 3*48 ints

    k_zero_counts<<<(Nn + 256) / 256, 256, 0, stream>>>(counts);

    for (int inp = 0; inp < 3; inp++) {
        const float* x = (const float*)d_in[inp];
        float* outR = out + 32 + (size_t)inp * Nn;             // r1/r2/r3 region
        unsigned long long* bk = bestkey + inp * 16;
        int* st = stats + inp * 48;

        k_cmean_wmma<<<4096, 256, 0, stream>>>(x, outR);
        k_minmax<<<16, 256, 0, stream>>>(outR, mnmx);
        k_norm_init<<<Nn / 256, 256, 0, stream>>>(outR, mnmx, parent);
        k_zero_small<<<1, 64, 0, stream>>>(bk, st);
        k_union<<<Nn / 256, 256, 0, stream>>>(parent);
        k_flatten<<<Nn / 256, 256, 0, stream>>>(parent);
        for (int b = 0; b < Bn; b++) {
            k_count<<<HWn / 256, 256, 0, stream>>>(parent, counts, b);
            k_argmax<<<HWn / 256, 256, 0, stream>>>(parent, counts, bk, b);
            k_reset<<<HWn / 256, 256, 0, stream>>>(parent, counts, b);
        }
        k_big<<<Nn / 256, 256, 0, stream>>>(parent, bk, outR, st);
    }
    k_final<<<1, 32, 0, stream>>>(stats, out, out + 16);
}